// SumSquaredError_32693291057282
// MI455X (gfx1250) — compile-verified
//
#include <hip/hip_runtime.h>

#define S_GRID 7
#define C_NUM 20
#define TILE_CELLS 256          // one cell per thread per tile
#define CELL_FLOATS 30          // 120 bytes per cell
#define TILE_FLOATS (TILE_CELLS * CELL_FLOATS)   // 7680 floats = 30720 B per tensor
#define TILE_CHUNKS (TILE_FLOATS / 2)            // 3840 b64 chunks = 15 per thread

typedef float v2f __attribute__((ext_vector_type(2)));
typedef float v8f __attribute__((ext_vector_type(8)));

__device__ __forceinline__ float wave_sum32(float v) {
  v += __shfl_xor(v, 16, 32);
  v += __shfl_xor(v, 8, 32);
  v += __shfl_xor(v, 4, 32);
  v += __shfl_xor(v, 2, 32);
  v += __shfl_xor(v, 1, 32);
  return v;
}

// Async global->LDS b64 copy (gfx1250). LDS operand is the wave-relative LDS
// byte address = flat shared address truncated to 32 bits.
__device__ __forceinline__ void async_copy_b64(uint32_t lds_off, const void* gaddr) {
  asm volatile("global_load_async_to_lds_b64 %0, %1, off"
               :: "v"(lds_off), "v"(gaddr)
               : "memory");
}

__device__ __forceinline__ void wait_asynccnt0() {
  asm volatile("s_wait_asynccnt 0x0" ::: "memory");
}

// Kernel 1: stage 256-cell tiles of gt/pred into LDS with fully coalesced
// async b64 copies, compute per-cell YOLO loss terms from LDS (conflict-free:
// 30-dword cell stride maps 32 lanes onto 32 distinct bank pairs), reduce to
// 6 partial sums per block. ws row layout: 16 floats (cols 0..5 used, 6..15
// zero) so kernel 2 maps WMMA B columns directly.
__global__ void __launch_bounds__(256) yolo_partial(const float* __restrict__ gt,
                                                    const float* __restrict__ pred,
                                                    float* __restrict__ ws,
                                                    int ncells) {
  __shared__ __align__(16) float sg[TILE_FLOATS];
  __shared__ __align__(16) float sp[TILE_FLOATS];

  float a0 = 0.f, a1 = 0.f, a2 = 0.f, a3 = 0.f, a4 = 0.f, a5 = 0.f;
  const int t = threadIdx.x;
  const int tile_stride = gridDim.x * TILE_CELLS;

  for (int base = blockIdx.x * TILE_CELLS; base < ncells; base += tile_stride) {
    const int tcells = min(TILE_CELLS, ncells - base);
    const int nch = tcells * (CELL_FLOATS / 2);          // b64 chunks per tensor
    const float* gtile = gt + (size_t)base * CELL_FLOATS;
    const float* ptile = pred + (size_t)base * CELL_FLOATS;

    // issue fully coalesced async copies: chunk k = i*256 + t is contiguous
#pragma unroll
    for (int i = 0; i < 15; ++i) {
      const int k = i * TILE_CELLS + t;
      if (k < nch) {
        async_copy_b64((uint32_t)(uintptr_t)&sg[2 * k], gtile + 2 * k);
        async_copy_b64((uint32_t)(uintptr_t)&sp[2 * k], ptile + 2 * k);
      }
    }
    // speculative prefetch of next tile region (global_prefetch_b8)
    __builtin_prefetch(gtile + (size_t)tile_stride * CELL_FLOATS + t * CELL_FLOATS, 0, 0);
    __builtin_prefetch(ptile + (size_t)tile_stride * CELL_FLOATS + t * CELL_FLOATS, 0, 0);

    wait_asynccnt0();       // this wave's copies landed in LDS
    __syncthreads();        // all waves' copies visible

    if (t < tcells) {
      const float2* g2 = reinterpret_cast<const float2*>(&sg[t * CELL_FLOATS]);
      const float2* p2 = reinterpret_cast<const float2*>(&sp[t * CELL_FLOATS]);
      float g[30], p[30];
#pragma unroll
      for (int j = 0; j < 15; ++j) {
        float2 a = g2[j]; g[2 * j] = a.x; g[2 * j + 1] = a.y;
        float2 b = p2[j]; p[2 * j] = b.x; p[2 * j + 1] = b.y;
      }

      // per-box elementwise IoU (center format)
      float iou[2];
#pragma unroll
      for (int b = 0; b < 2; ++b) {
        const float* a = &g[5 * b];
        const float* q = &p[5 * b];
        float ax1 = a[0] - a[2] * 0.5f, ay1 = a[1] - a[3] * 0.5f;
        float ax2 = a[0] + a[2] * 0.5f, ay2 = a[1] + a[3] * 0.5f;
        float bx1 = q[0] - q[2] * 0.5f, by1 = q[1] - q[3] * 0.5f;
        float bx2 = q[0] + q[2] * 0.5f, by2 = q[1] + q[3] * 0.5f;
        float iw = fmaxf(fminf(ax2, bx2) - fmaxf(ax1, bx1), 0.f);
        float ih = fmaxf(fminf(ay2, by2) - fmaxf(ay1, by1), 0.f);
        float inter = iw * ih;
        float uni = (ax2 - ax1) * (ay2 - ay1) + (bx2 - bx1) * (by2 - by1) - inter;
        iou[b] = inter / (uni + 1e-10f);
      }
      // jnp.argmax tie-break: first max wins -> box 1 only if strictly greater
      const int maxidx = (iou[1] > iou[0]) ? 1 : 0;
      const bool obj_i = (g[4] == 1.0f);

#pragma unroll
      for (int b = 0; b < 2; ++b) {
        const bool obj_ij = (g[4 + 5 * b] == 1.0f);
        const bool nonresp = obj_i && (b != maxidx);
        const bool m = obj_ij && !nonresp;
        const float gtc0 = nonresp ? 0.0f : g[4 + 5 * b];
        const float fm = m ? 1.0f : 0.0f;

        a0 += fm;                       // n_obj
        float bs = 0.f;
#pragma unroll
        for (int k = 0; k < 4; ++k) {
          float d = p[5 * b + k] - g[5 * b + k];
          bs += d * d;
        }
        a1 += fm * bs;                  // box SSE (masked)
        float dc = p[4 + 5 * b] - gtc0;
        a2 += fm * dc * dc;             // obj SSE
        a3 += (1.0f - fm) * dc * dc;    // noobj SSE
      }

      if (obj_i) {
        a4 += 1.0f;                     // n_cell
        float cs = 0.f;
#pragma unroll
        for (int j = 0; j < 20; ++j) {
          float d = p[10 + j] - g[10 + j];
          cs += d * d;
        }
        a5 += cs;                       // class SSE
      }
    }
    __syncthreads();        // LDS reads done before next tile overwrites
  }

  // wave32 reduction, then cross-wave via LDS (deterministic tree)
  a0 = wave_sum32(a0); a1 = wave_sum32(a1); a2 = wave_sum32(a2);
  a3 = wave_sum32(a3); a4 = wave_sum32(a4); a5 = wave_sum32(a5);

  __shared__ float smem[8][6];
  const int w = t >> 5, lane = t & 31;
  if (lane == 0) {
    smem[w][0] = a0; smem[w][1] = a1; smem[w][2] = a2;
    smem[w][3] = a3; smem[w][4] = a4; smem[w][5] = a5;
  }
  __syncthreads();
  if (t < 16) {
    float v = 0.f;
    if (t < 6) {
#pragma unroll
      for (int i = 0; i < 8; ++i) v += smem[i][t];
    }
    ws[(size_t)blockIdx.x * 16 + t] = v;   // cols 6..15 written as 0
  }
}

// Kernel 2: one wave. Column-sums of the nrows x 16 partial matrix via
// D = ones(16x4) x B(4x16) + 0 on v_wmma_f32_16x16x4_f32. A = all-ones is
// layout-invariant; summing over the full K dim makes the B slot->K mapping
// irrelevant (column n's 4 K-slots are {lane n, lane n+16} x {v0, v1}).
__global__ void yolo_final(const float* __restrict__ ws, float* __restrict__ out,
                           int nrows, int ncells) {
  const int L = threadIdx.x;
  const int h = L >> 4;       // lane-group: rows 2h, 2h+1 of each 4-row tile
  const int c = L & 15;       // column 0..15

  float v0 = 0.f, v1 = 0.f;
  for (int r = 0; r < nrows; r += 4) {   // uniform loop, EXEC stays full
    v0 += ws[(size_t)(r + 2 * h) * 16 + c];
    v1 += ws[(size_t)(r + 2 * h + 1) * 16 + c];
  }

  v2f A;  A[0] = 1.0f; A[1] = 1.0f;      // ones matrix: layout-invariant
  v2f Bm; Bm[0] = v0;  Bm[1] = v1;
  v8f C = {};
  v8f D = __builtin_amdgcn_wmma_f32_16x16x4_f32(
      /*neg_a=*/false, A, /*neg_b=*/false, Bm,
      /*c_mod=*/(short)0, C, /*reuse_a=*/false, /*reuse_b=*/false);

  // D row 0 lives in VGPR0, lanes 0..15 (N = lane); every row is identical.
  float tt = D[0];
  float t0 = __shfl(tt, 0, 32), t1 = __shfl(tt, 1, 32), t2 = __shfl(tt, 2, 32);
  float t3 = __shfl(tt, 3, 32), t4 = __shfl(tt, 4, 32), t5 = __shfl(tt, 5, 32);

  if (L == 0) {
    const float bz = (float)(ncells / (S_GRID * S_GRID));
    const float n_obj   = fmaxf(t0, 1.0f);
    const float n_noobj = fmaxf((float)ncells * 2.0f - t0, 1.0f);
    const float n_cell  = fmaxf(t4, 1.0f);
    const float box_loss   = t1 / (n_obj * 4.0f);
    const float obj_loss   = t2 / n_obj;
    const float noobj_loss = t3 / n_noobj;
    const float cls_loss   = t5 / (n_cell * (float)C_NUM);
    out[0] = 5.0f * box_loss / bz;                    // LAMBDA_COORD
    out[1] = (0.5f * noobj_loss + obj_loss) / bz;     // LAMBDA_NOOBJ
    out[2] = cls_loss / bz;
  }
}

extern "C" void kernel_launch(void* const* d_in, const int* in_sizes, int n_in,
                              void* d_out, int out_size, void* d_ws, size_t ws_size,
                              hipStream_t stream) {
  const float* gt = (const float*)d_in[0];
  const float* pred = (const float*)d_in[1];
  float* out = (float*)d_out;
  float* ws = (float*)d_ws;

  const int ncells = in_sizes[0] / 30;   // BZ * S * S

  int nb = 512;                          // 512 blocks x 256 thr, ~3 tiles/block
  const size_t need = (size_t)nb * 16 * sizeof(float);
  if (ws_size < need) {
    nb = (int)(ws_size / (16 * sizeof(float)));
    nb &= ~3;                            // kernel2 consumes rows in tiles of 4
    if (nb < 4) nb = 4;
  }

  yolo_partial<<<nb, 256, 0, stream>>>(gt, pred, ws, ncells);
  yolo_final<<<1, 32, 0, stream>>>(ws, out, nb, ncells);
}